// GAT_GCN_54314156425583
// MI455X (gfx1250) — compile-verified
//
#include <hip/hip_runtime.h>
#include <hip/hip_bf16.h>

// ---------------- constants from the reference ----------------
#define HEADS     10
#define F_H       35
#define F_IN      35
#define F_G       350     // HEADS * F_H
#define SEQL      42
#define EMBD      128
#define NFILT     16
#define KWIN      8
#define FCG1_DIM  1500
#define OUT_DIM   128
#define FC1_DIM   512
#define NEG_SLOPE 0.2f
#define NTILE     4       // 16-wide N tiles per wave (16x64 strip)

typedef float v2f __attribute__((ext_vector_type(2)));
typedef float v8f __attribute__((ext_vector_type(8)));

// ------------- helpers -------------
__device__ __forceinline__ unsigned f2key(float f) {
    unsigned b = __float_as_uint(f);
    return b ^ ((b >> 31) ? 0xFFFFFFFFu : 0x80000000u);
}
__device__ __forceinline__ float key2f(unsigned k) {
    if (k == 0u) return 0.0f; // untouched segment -> treat as 0 (matches isfinite guard)
    return (k & 0x80000000u) ? __uint_as_float(k ^ 0x80000000u)
                             : __uint_as_float(~k);
}
__device__ __forceinline__ void edge_sd(const int* __restrict__ ei, int e, int E,
                                        int& s, int& d) {
    if (e < E) { s = ei[e]; d = ei[E + e]; }
    else       { s = e - E; d = e - E; }     // self loops appended
}

// =====================================================================
// WMMA f32 GEMM: C[M,N] = act(A[M,K] @ B[K,N] + bias)
// One wave32 per 16x64 output strip: one A fragment is reused across 4
// V_WMMA_F32_16X16X4_F32 issues per K-step. Out-of-range rows/cols are
// CLAMPED for loads (their results are discarded at store time), so the
// hot loop has zero predication. All B fragments of a K-step are loaded
// BEFORE the 4 WMMAs so the loads coalesce into one clause / one wait
// and the WMMAs issue back-to-back on independent accumulators.
// =====================================================================
__global__ void k_wmma_gemm(const float* __restrict__ A, const float* __restrict__ B,
                            const float* __restrict__ bias, float* __restrict__ C,
                            int M, int N, int K, int lda, int ldb, int ldc, int act)
{
    const int wavesPerBlock = blockDim.x >> 5;
    const int wave   = blockIdx.x * wavesPerBlock + (threadIdx.x >> 5);
    const int tilesM  = (M + 15) >> 4;
    const int stripsN = (N + 63) >> 6;
    if (wave >= tilesM * stripsN) return;
    const int tm = wave / stripsN;
    const int ts = wave % stripsN;
    const int lane = threadIdx.x & 31;
    const int half = lane >> 4;      // 0: lanes 0-15, 1: lanes 16-31
    const int l16  = lane & 15;

    const int row  = tm << 4;
    const int col0 = ts << 6;

    // clamped load coordinates (discarded at store when out of range)
    const int ar = min(row + l16, M - 1);
    int bc[NTILE];
    #pragma unroll
    for (int t = 0; t < NTILE; ++t) bc[t] = min(col0 + 16 * t + l16, N - 1);

    v8f acc[NTILE] = {};

    // k0 = kb + 2*half per the ISA 16x4 f32 A/B lane layout
    const float* Ap = A + (size_t)ar * lda + 2 * half;
    const float* Bp = B + (size_t)(2 * half) * ldb;

    const int Kmain = K & ~3;
    for (int kb = 0; kb < Kmain; kb += 4) {
        v2f a;
        a.x = Ap[0];
        a.y = Ap[1];
        v2f b[NTILE];
        #pragma unroll
        for (int t = 0; t < NTILE; ++t) {          // batch all loads first
            b[t].x = Bp[bc[t]];
            b[t].y = Bp[(size_t)ldb + bc[t]];
        }
        #pragma unroll
        for (int t = 0; t < NTILE; ++t) {          // then back-to-back WMMAs
            acc[t] = __builtin_amdgcn_wmma_f32_16x16x4_f32(
                         false, a, false, b[t], (short)0, acc[t], false, false);
        }
        Ap += 4;
        Bp += (size_t)4 * ldb;
    }
    if (Kmain < K) {                       // K-tail, executed at most once
        const int k0 = Kmain + 2 * half;
        v2f a = {0.f, 0.f};
        if (k0     < K) a.x = A[(size_t)ar * lda + k0];
        if (k0 + 1 < K) a.y = A[(size_t)ar * lda + k0 + 1];
        v2f b[NTILE];
        #pragma unroll
        for (int t = 0; t < NTILE; ++t) {
            b[t].x = 0.f; b[t].y = 0.f;
            if (k0     < K) b[t].x = B[(size_t)k0 * ldb + bc[t]];
            if (k0 + 1 < K) b[t].y = B[(size_t)(k0 + 1) * ldb + bc[t]];
        }
        #pragma unroll
        for (int t = 0; t < NTILE; ++t) {
            acc[t] = __builtin_amdgcn_wmma_f32_16x16x4_f32(
                         false, a, false, b[t], (short)0, acc[t], false, false);
        }
    }

    #pragma unroll
    for (int t = 0; t < NTILE; ++t) {
        const int n = col0 + 16 * t + l16;
        if (n >= N) continue;
        const float bv = bias ? bias[n] : 0.0f;
        #pragma unroll
        for (int r = 0; r < 8; ++r) {
            const int m = row + r + 8 * half;   // C/D layout: VGPR r -> rows r, r+8
            if (m < M) {
                float v = acc[t][r] + bv;
                if (act == 1) v = fmaxf(v, 0.0f);
                C[(size_t)m * ldc + n] = v;
            }
        }
    }
}

// =====================================================================
// GAT attention pipeline
// =====================================================================
__global__ void k_node_att(const float* __restrict__ H,
                           const float* __restrict__ att_src,
                           const float* __restrict__ att_dst,
                           float* __restrict__ AS, float* __restrict__ AD, int N)
{
    int i = blockIdx.x * blockDim.x + threadIdx.x;
    if (i >= N * HEADS) return;
    int n = i / HEADS, h = i % HEADS;
    const float* hp = H + (size_t)n * F_G + h * F_H;
    const float* s  = att_src + h * F_H;
    const float* d  = att_dst + h * F_H;
    float as = 0.f, ad = 0.f;
    #pragma unroll
    for (int c = 0; c < F_H; ++c) { float v = hp[c]; as += v * s[c]; ad += v * d[c]; }
    AS[i] = as; AD[i] = ad;
}

__global__ void k_edge_max(const int* __restrict__ ei,
                           const float* __restrict__ AS, const float* __restrict__ AD,
                           unsigned* __restrict__ mkey, int E, int Etot)
{
    int i = blockIdx.x * blockDim.x + threadIdx.x;
    if (i >= Etot * HEADS) return;
    int e = i / HEADS, h = i % HEADS;
    int s, d; edge_sd(ei, e, E, s, d);
    float x = AS[s * HEADS + h] + AD[d * HEADS + h];
    float v = (x > 0.f) ? x : NEG_SLOPE * x;            // leaky relu
    atomicMax(&mkey[d * HEADS + h], f2key(v));
}

__global__ void k_edge_exp(const int* __restrict__ ei,
                           const float* __restrict__ AS, const float* __restrict__ AD,
                           const unsigned* __restrict__ mkey,
                           float* __restrict__ den, float* __restrict__ alpha,
                           int E, int Etot)
{
    int i = blockIdx.x * blockDim.x + threadIdx.x;
    if (i >= Etot * HEADS) return;
    int e = i / HEADS, h = i % HEADS;
    int s, d; edge_sd(ei, e, E, s, d);
    float x = AS[s * HEADS + h] + AD[d * HEADS + h];
    float v = (x > 0.f) ? x : NEG_SLOPE * x;
    float m = key2f(mkey[d * HEADS + h]);
    float ex = __expf(v - m);
    atomicAdd(&den[d * HEADS + h], ex);
    alpha[i] = ex;
}

__global__ void k_edge_norm(const int* __restrict__ ei,
                            const float* __restrict__ den,
                            float* __restrict__ alpha, int E, int Etot)
{
    int i = blockIdx.x * blockDim.x + threadIdx.x;
    if (i >= Etot * HEADS) return;
    int e = i / HEADS, h = i % HEADS;
    int s, d; edge_sd(ei, e, E, s, d);
    alpha[i] = alpha[i] / (den[d * HEADS + h] + 1e-16f);
}

__global__ void k_edge_agg(const int* __restrict__ ei, const float* __restrict__ H,
                           const float* __restrict__ alpha, float* __restrict__ XG,
                           int E, int Etot)
{
    int i = blockIdx.x * blockDim.x + threadIdx.x;
    if (i >= Etot * HEADS) return;
    int e = i / HEADS, h = i % HEADS;
    int s, d; edge_sd(ei, e, E, s, d);
    float a = alpha[i];
    const float* hp = H + (size_t)s * F_G + h * F_H;
    float* op = XG + (size_t)d * F_G + h * F_H;
    #pragma unroll
    for (int c = 0; c < F_H; ++c) atomicAdd(&op[c], hp[c] * a);
}

__global__ void k_bias_relu(float* __restrict__ X, const float* __restrict__ bias,
                            long long total, int ld)
{
    long long i = (long long)blockIdx.x * blockDim.x + threadIdx.x;
    if (i >= total) return;
    int j = (int)(i % ld);
    X[i] = fmaxf(X[i] + bias[j], 0.0f);
}

// =====================================================================
// GCN pipeline
// =====================================================================
__global__ void k_deg(const int* __restrict__ ei, float* __restrict__ deg,
                      int E, int Etot)
{
    int e = blockIdx.x * blockDim.x + threadIdx.x;
    if (e >= Etot) return;
    int s, d; edge_sd(ei, e, E, s, d);
    atomicAdd(&deg[d], 1.0f);
}

__global__ void k_dinv(const float* __restrict__ deg, float* __restrict__ dinv, int N)
{
    int n = blockIdx.x * blockDim.x + threadIdx.x;
    if (n >= N) return;
    float dg = deg[n];
    dinv[n] = (dg > 0.f) ? rsqrtf(dg) : 0.0f;
}

__global__ void k_gcn_scatter(const int* __restrict__ ei, const float* __restrict__ H2,
                              const float* __restrict__ dinv, float* __restrict__ XG,
                              int E, int Etot)
{
    long long total = (long long)Etot * F_G;
    long long i = (long long)blockIdx.x * blockDim.x + threadIdx.x;
    if (i >= total) return;
    int e = (int)(i / F_G);
    int j = (int)(i % F_G);
    int s, d; edge_sd(ei, e, E, s, d);
    float nrm = dinv[s] * dinv[d];
    atomicAdd(&XG[(size_t)d * F_G + j], H2[(size_t)s * F_G + j] * nrm);
}

// =====================================================================
// Global pooling (max + mean per graph)
// =====================================================================
__global__ void k_pool(const float* __restrict__ XG, const int* __restrict__ batch,
                       unsigned* __restrict__ gmaxk, float* __restrict__ gsum,
                       float* __restrict__ cnt, int N)
{
    long long total = (long long)N * F_G;
    long long i = (long long)blockIdx.x * blockDim.x + threadIdx.x;
    if (i >= total) return;
    int n = (int)(i / F_G);
    int j = (int)(i % F_G);
    int b = batch[n];
    float v = XG[i];
    atomicMax(&gmaxk[b * F_G + j], f2key(v));
    atomicAdd(&gsum[b * F_G + j], v);
    if (j == 0) atomicAdd(&cnt[b], 1.0f);
}

__global__ void k_pool_combine(const unsigned* __restrict__ gmaxk,
                               const float* __restrict__ gsum,
                               const float* __restrict__ cnt,
                               float* __restrict__ pooled, int B)
{
    int i = blockIdx.x * blockDim.x + threadIdx.x;
    if (i >= B * F_G) return;
    int b = i / F_G, j = i % F_G;
    pooled[(size_t)b * (2 * F_G) + j]       = key2f(gmaxk[i]);
    pooled[(size_t)b * (2 * F_G) + F_G + j] = gsum[i] / fmaxf(cnt[b], 1.0f);
}

// =====================================================================
// Protein branch: embedding gather + 1D conv (VALID), K=8, 128 channels
// =====================================================================
__global__ void k_conv1d(const int* __restrict__ target, const float* __restrict__ emb,
                         const float* __restrict__ Wc, const float* __restrict__ bc,
                         float* __restrict__ conv, int B)
{
    const int TOUT = SEQL - KWIN + 1; // 35
    int i = blockIdx.x * blockDim.x + threadIdx.x;
    if (i >= B * NFILT * TOUT) return;
    int t = i % TOUT;
    int f = (i / TOUT) % NFILT;
    int b = i / (TOUT * NFILT);
    float acc = 0.0f;
    #pragma unroll
    for (int k = 0; k < KWIN; ++k) {
        const float* ev = emb + (size_t)target[b * SEQL + t + k] * EMBD;
        const float* wv = Wc  + (size_t)f * EMBD * KWIN + k;   // W[f][c][k]
        for (int c = 0; c < EMBD; ++c) acc += ev[c] * wv[(size_t)c * KWIN];
    }
    conv[(size_t)b * (NFILT * TOUT) + f * TOUT + t] = acc + bc[f];
}

// =====================================================================
// Final row-dot + sigmoid: out[b] = sigmoid(F1[b,:] . W_out + b_out)
// =====================================================================
__global__ void k_out(const float* __restrict__ F1, const float* __restrict__ Wout,
                      const float* __restrict__ bout, float* __restrict__ out, int B)
{
    int b = blockIdx.x * blockDim.x + threadIdx.x;
    if (b >= B) return;
    float acc = 0.0f;
    const float* r = F1 + (size_t)b * FC1_DIM;
    for (int j = 0; j < FC1_DIM; ++j) acc += r[j] * Wout[j];
    float z = acc + bout[0];
    out[b] = 1.0f / (1.0f + __expf(-z));
}

// =====================================================================
// host orchestration
// =====================================================================
static inline int cdiv(long long a, int b) { return (int)((a + b - 1) / b); }

extern "C" void kernel_launch(void* const* d_in, const int* in_sizes, int n_in,
                              void* d_out, int out_size, void* d_ws, size_t ws_size,
                              hipStream_t stream)
{
    // ---- inputs (setup_inputs order) ----
    const float* x       = (const float*)d_in[0];
    const int*   ei      = (const int*)  d_in[1];
    const int*   batch   = (const int*)  d_in[2];
    const int*   target  = (const int*)  d_in[3];
    const float* W_gat   = (const float*)d_in[4];
    const float* att_src = (const float*)d_in[5];
    const float* att_dst = (const float*)d_in[6];
    const float* b_gat   = (const float*)d_in[7];
    const float* W_gcn   = (const float*)d_in[8];
    const float* b_gcn   = (const float*)d_in[9];
    const float* W_fcg1  = (const float*)d_in[10];
    const float* b_fcg1  = (const float*)d_in[11];
    const float* W_fcg2  = (const float*)d_in[12];
    const float* b_fcg2  = (const float*)d_in[13];
    const float* emb_xt  = (const float*)d_in[14];
    const float* W_conv  = (const float*)d_in[15];
    const float* b_conv  = (const float*)d_in[16];
    const float* W_fcxt  = (const float*)d_in[17];
    const float* b_fcxt  = (const float*)d_in[18];
    const float* W_fc1   = (const float*)d_in[19];
    const float* b_fc1   = (const float*)d_in[20];
    const float* W_out   = (const float*)d_in[21];
    const float* b_out   = (const float*)d_in[22];

    const int N    = in_sizes[0] / F_IN;
    const int E    = in_sizes[1] / 2;
    const int B    = in_sizes[3] / SEQL;
    const int Etot = E + N;

    // ---- outputs: (out[B,1], alpha[Etot,HEADS]) concatenated ----
    float* out_v = (float*)d_out;
    float* alpha = out_v + B;

    // ---- workspace bump allocator ----
    char* ws = (char*)d_ws;
    size_t off = 0;
    auto alloc = [&](size_t bytes) -> void* {
        void* p = ws + off;
        off = (off + bytes + 255) & ~(size_t)255;
        return p;
    };
    float*    H     = (float*)   alloc((size_t)N * F_G * 4);     // GAT h, reused for GCN GEMM
    float*    XG    = (float*)   alloc((size_t)N * F_G * 4);     // node features, reused
    float*    AS    = (float*)   alloc((size_t)N * HEADS * 4);
    float*    AD    = (float*)   alloc((size_t)N * HEADS * 4);
    unsigned* MKEY  = (unsigned*)alloc((size_t)N * HEADS * 4);
    float*    DEN   = (float*)   alloc((size_t)N * HEADS * 4);
    float*    DEG   = (float*)   alloc((size_t)N * 4);
    float*    DINV  = (float*)   alloc((size_t)N * 4);
    unsigned* GMAXK = (unsigned*)alloc((size_t)B * F_G * 4);
    float*    GSUM  = (float*)   alloc((size_t)B * F_G * 4);
    float*    CNT   = (float*)   alloc((size_t)B * 4);
    float*    POOL  = (float*)   alloc((size_t)B * 2 * F_G * 4);
    float*    FG1   = (float*)   alloc((size_t)B * FCG1_DIM * 4);
    float*    XC    = (float*)   alloc((size_t)B * 256 * 4);     // [xg(128) | xt(128)]
    float*    CONV  = (float*)   alloc((size_t)B * NFILT * 35 * 4);
    float*    F1    = (float*)   alloc((size_t)B * FC1_DIM * 4);
    (void)ws_size;

    // ---- zero-init accumulators (key=0 encodes "-inf" for the max trick) ----
    hipMemsetAsync(MKEY,  0, (size_t)N * HEADS * 4, stream);
    hipMemsetAsync(DEN,   0, (size_t)N * HEADS * 4, stream);
    hipMemsetAsync(DEG,   0, (size_t)N * 4,         stream);
    hipMemsetAsync(GMAXK, 0, (size_t)B * F_G * 4,   stream);
    hipMemsetAsync(GSUM,  0, (size_t)B * F_G * 4,   stream);
    hipMemsetAsync(CNT,   0, (size_t)B * 4,         stream);
    hipMemsetAsync(XG,    0, (size_t)N * F_G * 4,   stream);

    const int TB = 256;
    auto gemm = [&](const float* A, const float* Bm, const float* bias, float* C,
                    int M, int Nn, int K, int lda, int ldb, int ldc, int act) {
        long long tiles = (long long)((M + 15) / 16) * ((Nn + 63) / 64);
        int blocks = cdiv(tiles, 4);           // 4 waves (16x64 strips) per 128-thread block
        k_wmma_gemm<<<blocks, 128, 0, stream>>>(A, Bm, bias, C, M, Nn, K,
                                                lda, ldb, ldc, act);
    };

    // ---- 1. GAT projection: H = x @ W_gat (no bias yet) ----
    gemm(x, W_gat, nullptr, H, N, F_G, F_IN, F_IN, F_G, F_G, 0);

    // ---- 2. attention logits / segment softmax / aggregation ----
    k_node_att<<<cdiv((long long)N * HEADS, TB), TB, 0, stream>>>(H, att_src, att_dst, AS, AD, N);
    k_edge_max<<<cdiv((long long)Etot * HEADS, TB), TB, 0, stream>>>(ei, AS, AD, MKEY, E, Etot);
    k_edge_exp<<<cdiv((long long)Etot * HEADS, TB), TB, 0, stream>>>(ei, AS, AD, MKEY, DEN, alpha, E, Etot);
    k_edge_norm<<<cdiv((long long)Etot * HEADS, TB), TB, 0, stream>>>(ei, DEN, alpha, E, Etot);
    k_edge_agg<<<cdiv((long long)Etot * HEADS, TB), TB, 0, stream>>>(ei, H, alpha, XG, E, Etot);
    k_bias_relu<<<cdiv((long long)N * F_G, TB), TB, 0, stream>>>(XG, b_gat, (long long)N * F_G, F_G);

    // ---- 3. GCN: deg norm, GEMM (reuse H), scatter back into XG ----
    k_deg<<<cdiv(Etot, TB), TB, 0, stream>>>(ei, DEG, E, Etot);
    k_dinv<<<cdiv(N, TB), TB, 0, stream>>>(DEG, DINV, N);
    gemm(XG, W_gcn, nullptr, H, N, F_G, F_G, F_G, F_G, F_G, 0);
    hipMemsetAsync(XG, 0, (size_t)N * F_G * 4, stream);
    k_gcn_scatter<<<cdiv((long long)Etot * F_G, TB), TB, 0, stream>>>(ei, H, DINV, XG, E, Etot);
    k_bias_relu<<<cdiv((long long)N * F_G, TB), TB, 0, stream>>>(XG, b_gcn, (long long)N * F_G, F_G);

    // ---- 4. pooling ----
    k_pool<<<cdiv((long long)N * F_G, TB), TB, 0, stream>>>(XG, batch, GMAXK, GSUM, CNT, N);
    k_pool_combine<<<cdiv((long long)B * F_G, TB), TB, 0, stream>>>(GMAXK, GSUM, CNT, POOL, B);

    // ---- 5. graph FC stack ----
    gemm(POOL, W_fcg1, b_fcg1, FG1, B, FCG1_DIM, 2 * F_G, 2 * F_G, FCG1_DIM, FCG1_DIM, 1);
    gemm(FG1, W_fcg2, b_fcg2, XC, B, OUT_DIM, FCG1_DIM, FCG1_DIM, OUT_DIM, 256, 0); // -> XC[:, :128]

    // ---- 6. protein branch ----
    k_conv1d<<<cdiv((long long)B * NFILT * 35, TB), TB, 0, stream>>>(target, emb_xt, W_conv, b_conv, CONV, B);
    gemm(CONV, W_fcxt, b_fcxt, XC + OUT_DIM, B, OUT_DIM, NFILT * 35, NFILT * 35, OUT_DIM, 256, 0); // -> XC[:, 128:]

    // ---- 7. fusion ----
    gemm(XC, W_fc1, b_fc1, F1, B, FC1_DIM, 256, 256, FC1_DIM, FC1_DIM, 1);
    k_out<<<cdiv(B, 128), 128, 0, stream>>>(F1, W_out, b_out, out_v, B);
}